// _GIN_59313498358208
// MI455X (gfx1250) — compile-verified
//
#include <hip/hip_runtime.h>
#include <hip/hip_bf16.h>

#define DDIM 64
#define WT_STRIDE 66        // padded transposed-W row stride (bank-conflict-free)
#define TILES_PER_BLOCK 8   // 8 waves * 16 rows = 128 rows per block

typedef __attribute__((ext_vector_type(2))) float v2f;
typedef __attribute__((ext_vector_type(8))) float v8f;

// ---------------------------------------------------------------------------
// float4 copy: dst = src  (n4 = element count / 4)
// ---------------------------------------------------------------------------
__global__ void gin_copy_kernel(const float* __restrict__ src,
                                float* __restrict__ dst, int n4) {
    int i = blockIdx.x * blockDim.x + threadIdx.x;
    if (i < n4) {
        reinterpret_cast<float4*>(dst)[i] =
            reinterpret_cast<const float4*>(src)[i];
    }
}

// ---------------------------------------------------------------------------
// Edge scatter-add: agg[dst[e]] += h[src[e]]  (agg pre-initialized with h,
// folding the GIN "(1+eps)*h + sum" with eps=0).
// 16 threads per edge, 4 consecutive floats each -> 256B coalesced per edge.
// All of h + edge lists (~20 MB) are L2-resident on MI455X (192 MB L2), so
// the f32 atomics resolve in L2.
// ---------------------------------------------------------------------------
__global__ void gin_scatter_kernel(const float* __restrict__ h,
                                   const int* __restrict__ esrc,
                                   const int* __restrict__ edst,
                                   float* __restrict__ agg, int nedges) {
    int tid = blockIdx.x * blockDim.x + threadIdx.x;
    int e = tid >> 4;
    if (e >= nedges) return;
    int c = (tid & 15) << 2;
    int s = esrc[e];
    int d = edst[e];
    float4 v = *reinterpret_cast<const float4*>(h + (size_t)s * DDIM + c);
    float* p = agg + (size_t)d * DDIM + c;
    atomicAdd(p + 0, v.x);
    atomicAdd(p + 1, v.y);
    atomicAdd(p + 2, v.z);
    atomicAdd(p + 3, v.w);
}

// ---------------------------------------------------------------------------
// WMMA f32 GEMM:  Out[nrows x 64] = preop(A) @ W[64 x 64] + bias
// preop==1: per-column relu(a*scale[c]+shift[c]) fused into the A-load
// (this is the BN1+ReLU between the two linear layers).
//
// Uses V_WMMA_F32_16X16X4_F32, K=4 per issue, 16 issues per 16x16 tile.
// A frag (16x4 f32): lane L holds M=L%16; koff = (L>=16)?2:0, VGPR0/1 = K
// koff/koff+1 -> contiguous float2 per lane (global load b64).
// B frag (4x16 f32): VGPR r = row K = r + koff, N = n0 + L%16. W is staged
// TRANSPOSED in LDS with a padded stride so {W[kk][n], W[kk+1][n]} is one
// contiguous, bank-conflict-free ds_load_b64 into an even VGPR pair.
// C/D (16x16 f32, 8 VGPRs): VGPR r -> M = r + (L>=16 ? 8 : 0), N = n0 + L%16.
// ---------------------------------------------------------------------------
__global__ __launch_bounds__(256) void gin_gemm_wmma_kernel(
    const float* __restrict__ A, const float* __restrict__ W,
    const float* __restrict__ bias,
    const float* __restrict__ scale, const float* __restrict__ shift,
    float* __restrict__ Out, int ntiles, int preop) {
    __shared__ float sWt[DDIM * WT_STRIDE];   // sWt[n*WT_STRIDE + k] = W[k][n]
    __shared__ float sBias[DDIM];
    __shared__ float sScale[DDIM];
    __shared__ float sShift[DDIM];

    for (int i = threadIdx.x; i < DDIM * DDIM; i += 256) {
        int k = i >> 6;          // row of W (input dim)
        int n = i & 63;          // col of W (output dim)
        sWt[n * WT_STRIDE + k] = W[i];
    }
    if (threadIdx.x < DDIM) {
        sBias[threadIdx.x]  = bias[threadIdx.x];
        sScale[threadIdx.x] = preop ? scale[threadIdx.x] : 1.0f;
        sShift[threadIdx.x] = preop ? shift[threadIdx.x] : 0.0f;
    }
    __syncthreads();

    int lane = threadIdx.x & 31;
    int wave = threadIdx.x >> 5;
    int tile = blockIdx.x * TILES_PER_BLOCK + wave;
    if (tile >= ntiles) return;           // wave-uniform: EXEC stays all-1s

    int m    = lane & 15;
    int koff = (lane >> 4) << 1;          // 0 or 2

    // Load the 16x64 A stripe for this wave as 16 K-step fragments.
    const float* arow = A + ((size_t)tile * 16 + m) * DDIM;
    v2f a[16];
#pragma unroll
    for (int k = 0; k < 16; ++k) {
        int c0 = 4 * k + koff;
        float x0 = arow[c0];
        float x1 = arow[c0 + 1];
        if (preop) {
            x0 = fmaxf(fmaf(x0, sScale[c0],     sShift[c0]),     0.0f);
            x1 = fmaxf(fmaf(x1, sScale[c0 + 1], sShift[c0 + 1]), 0.0f);
        }
        a[k].x = x0;
        a[k].y = x1;
    }

#pragma unroll
    for (int nt = 0; nt < 4; ++nt) {
        int n = nt * 16 + m;
        const float* wcol = &sWt[n * WT_STRIDE + koff];
        v8f c = {};
#pragma unroll
        for (int k = 0; k < 16; ++k) {
            // contiguous pair {W[4k+koff][n], W[4k+koff+1][n]} -> ds_load_b64
            v2f b = *reinterpret_cast<const v2f*>(wcol + 4 * k);
            // 8 args: (neg_a, A, neg_b, B, c_mod, C, reuse_a, reuse_b)
            c = __builtin_amdgcn_wmma_f32_16x16x4_f32(
                false, a[k], false, b, (short)0, c, false, false);
        }
        float bb = sBias[n];
        int rbase = tile * 16 + ((lane >> 4) << 3);
#pragma unroll
        for (int r = 0; r < 8; ++r)
            Out[((size_t)(rbase + r)) * DDIM + n] = c[r] + bb;
    }
}

// ---------------------------------------------------------------------------
// Zero small stats region
// ---------------------------------------------------------------------------
__global__ void gin_zero_kernel(float* __restrict__ p, int n) {
    int i = blockIdx.x * blockDim.x + threadIdx.x;
    if (i < n) p[i] = 0.0f;
}

// ---------------------------------------------------------------------------
// Per-column sum / sum-of-squares over nrows (training-mode BN stats).
// preop==1: accumulate stats of relu(x*scale[c]+shift[c]) instead of x
// (used to get the bno stats of relu(bna(x)) without materializing it).
// ---------------------------------------------------------------------------
__global__ __launch_bounds__(256) void gin_bnstats_kernel(
    const float* __restrict__ x,
    const float* __restrict__ scale, const float* __restrict__ shift,
    float* __restrict__ sums, float* __restrict__ sq, int nrows, int preop) {
    __shared__ float ssum[256];
    __shared__ float ssq[256];
    int c   = threadIdx.x & 63;
    int grp = threadIdx.x >> 6;                 // 0..3 row streams per block
    int nstreams = gridDim.x * 4;
    int row0 = blockIdx.x * 4 + grp;
    float sc = preop ? scale[c] : 1.0f;
    float sh = preop ? shift[c] : 0.0f;
    float s = 0.0f, q = 0.0f;
    for (int r = row0; r < nrows; r += nstreams) {
        float v = x[(size_t)r * DDIM + c];
        if (preop) v = fmaxf(fmaf(v, sc, sh), 0.0f);
        s += v;
        q += v * v;
    }
    ssum[threadIdx.x] = s;
    ssq[threadIdx.x]  = q;
    __syncthreads();
    if (threadIdx.x < 64) {
        s = ssum[threadIdx.x] + ssum[threadIdx.x + 64] +
            ssum[threadIdx.x + 128] + ssum[threadIdx.x + 192];
        q = ssq[threadIdx.x] + ssq[threadIdx.x + 64] +
            ssq[threadIdx.x + 128] + ssq[threadIdx.x + 192];
        atomicAdd(&sums[threadIdx.x], s);
        atomicAdd(&sq[threadIdx.x], q);
    }
}

// ---------------------------------------------------------------------------
// Fold BN stats into affine:  scale = g*rsqrt(var+eps), shift = b - mean*scale
// (biased variance, matching jnp.var / torch normalization). 64 threads.
// ---------------------------------------------------------------------------
__global__ void gin_bnfinalize_kernel(const float* __restrict__ sums,
                                      const float* __restrict__ sq,
                                      const float* __restrict__ g,
                                      const float* __restrict__ b,
                                      float* __restrict__ scale,
                                      float* __restrict__ shift, float inv_n) {
    int c = threadIdx.x;
    float m   = sums[c] * inv_n;
    float var = sq[c] * inv_n - m * m;
    float sc  = g[c] * rsqrtf(var + 1e-5f);
    scale[c] = sc;
    shift[c] = b[c] - m * sc;
}

// ---------------------------------------------------------------------------
// out = relu( relu(x*sa+ba) * so + bo )   (bna then bno, both with ReLU)
// ---------------------------------------------------------------------------
__global__ void gin_finalact_kernel(const float* __restrict__ x,
                                    const float* __restrict__ sa,
                                    const float* __restrict__ ba,
                                    const float* __restrict__ so,
                                    const float* __restrict__ bo,
                                    float* __restrict__ out, int n) {
    int i = blockIdx.x * blockDim.x + threadIdx.x;
    if (i < n) {
        int c = i & 63;
        float t = fmaxf(fmaf(x[i], sa[c], ba[c]), 0.0f);
        out[i] = fmaxf(fmaf(t, so[c], bo[c]), 0.0f);
    }
}

// ---------------------------------------------------------------------------
extern "C" void kernel_launch(void* const* d_in, const int* in_sizes, int n_in,
                              void* d_out, int out_size, void* d_ws, size_t ws_size,
                              hipStream_t stream) {
    const float* feat  = (const float*)d_in[0];
    const float* W1    = (const float*)d_in[1];
    const float* b1    = (const float*)d_in[2];
    const float* bn1_g = (const float*)d_in[3];
    const float* bn1_b = (const float*)d_in[4];
    const float* W2    = (const float*)d_in[5];
    const float* b2    = (const float*)d_in[6];
    const float* bna_g = (const float*)d_in[7];
    const float* bna_b = (const float*)d_in[8];
    const float* bno_g = (const float*)d_in[9];
    const float* bno_b = (const float*)d_in[10];
    const int*   esrc  = (const int*)d_in[11];
    const int*   edst  = (const int*)d_in[12];

    const int nnodes = in_sizes[0] / DDIM;
    const int nedges = in_sizes[11];
    const int nelem  = nnodes * DDIM;
    const int n4     = nelem / 4;
    const int ntiles = (nnodes + 15) / 16;
    float* out = (float*)d_out;

    // workspace layout: [tmp (agg, later x2)] [x1] [stats 512 floats]
    char* ws = (char*)d_ws;
    size_t nb = ((size_t)nelem * sizeof(float) + 255) & ~(size_t)255;
    float* tmp = (float*)(ws);
    float* x1  = (float*)(ws + nb);
    float* st  = (float*)(ws + 2 * nb);
    float* sums   = st;        float* sq     = st + 64;
    float* scale1 = st + 128;  float* shift1 = st + 192;
    float* sa     = st + 256;  float* ba     = st + 320;
    float* so     = st + 384;  float* bo     = st + 448;

    const float inv_n = 1.0f / (float)nnodes;
    dim3 blk(256);
    dim3 gCopy((n4 + 255) / 256);
    dim3 gScat(((size_t)nedges * 16 + 255) / 256);
    dim3 gGemm((ntiles + TILES_PER_BLOCK - 1) / TILES_PER_BLOCK);
    dim3 gElem((nelem + 255) / 256);
    dim3 gStats(128);

    // output layer 0 = input features
    gin_copy_kernel<<<gCopy, blk, 0, stream>>>(feat, out, n4);

    const float* h = out;  // current layer input lives in d_out slice
    for (int l = 0; l < 2; ++l) {
        const float* W1l = W1 + (size_t)l * DDIM * DDIM;
        const float* b1l = b1 + l * DDIM;
        const float* W2l = W2 + (size_t)l * DDIM * DDIM;
        const float* b2l = b2 + l * DDIM;

        // agg = h + segment_sum(h[src] -> dst)
        gin_copy_kernel<<<gCopy, blk, 0, stream>>>(h, tmp, n4);
        gin_scatter_kernel<<<gScat, blk, 0, stream>>>(h, esrc, edst, tmp, nedges);

        // x1 = agg @ W1 + b1
        gin_gemm_wmma_kernel<<<gGemm, blk, 0, stream>>>(
            tmp, W1l, b1l, nullptr, nullptr, x1, ntiles, 0);

        // BN1 stats -> (scale1, shift1)
        gin_zero_kernel<<<1, 128, 0, stream>>>(sums, 128);
        gin_bnstats_kernel<<<gStats, blk, 0, stream>>>(
            x1, nullptr, nullptr, sums, sq, nnodes, 0);
        gin_bnfinalize_kernel<<<1, 64, 0, stream>>>(
            sums, sq, bn1_g + l * DDIM, bn1_b + l * DDIM, scale1, shift1, inv_n);

        // x2 = relu(bn1(x1)) @ W2 + b2   (BN+ReLU fused into A-load); x2 reuses tmp
        gin_gemm_wmma_kernel<<<gGemm, blk, 0, stream>>>(
            x1, W2l, b2l, scale1, shift1, tmp, ntiles, 1);

        // BN-a stats of x2 -> (sa, ba)
        gin_zero_kernel<<<1, 128, 0, stream>>>(sums, 128);
        gin_bnstats_kernel<<<gStats, blk, 0, stream>>>(
            tmp, nullptr, nullptr, sums, sq, nnodes, 0);
        gin_bnfinalize_kernel<<<1, 64, 0, stream>>>(
            sums, sq, bna_g + l * DDIM, bna_b + l * DDIM, sa, ba, inv_n);

        // BN-o stats of relu(bna(x2)) -> (so, bo), computed on the fly
        gin_zero_kernel<<<1, 128, 0, stream>>>(sums, 128);
        gin_bnstats_kernel<<<gStats, blk, 0, stream>>>(
            tmp, sa, ba, sums, sq, nnodes, 1);
        gin_bnfinalize_kernel<<<1, 64, 0, stream>>>(
            sums, sq, bno_g + l * DDIM, bno_b + l * DDIM, so, bo, inv_n);

        // h_next = relu(bno(relu(bna(x2)))) -> output slice l+1
        float* hout = out + (size_t)(l + 1) * nelem;
        gin_finalact_kernel<<<gElem, blk, 0, stream>>>(tmp, sa, ba, so, bo, hout, nelem);
        h = hout;
    }
}